// Model_39676907885720
// MI455X (gfx1250) — compile-verified
//
#include <hip/hip_runtime.h>
#include <hip/hip_bf16.h>

// ---------------------------------------------------------------------------
// CDNA5 (gfx1250) fp32 WMMA implementation of the reference attention-ish block.
// B=4, C=8, S=D=H=512  ->  32 independent 512x512 problems, 13 fused GEMMs.
// Matrix math on V_WMMA_F32_16X16X4_F32 (fp32 matches reference numerics).
// Operand staging via GLOBAL_LOAD_ASYNC_TO_LDS_B128 + double-buffered LDS,
// synchronized with s_wait_asynccnt (ASYNCcnt) — the CDNA5 async copy path.
// ---------------------------------------------------------------------------

typedef float v2f __attribute__((ext_vector_type(2)));
typedef float v8f __attribute__((ext_vector_type(8)));

#define NMAT 512
#define MATELEMS (512 * 512)           // elements per batch matrix
#define NBATCH 32                      // B*C
#define LDSTRIDE 20                    // floats; 80B row: 16B-aligned, conflict-free

// Async copy 16B global -> LDS (per lane). dst must be 16B-aligned LDS.
__device__ __forceinline__ void async_b128(void* lds_dst, const void* gsrc)
{
    // low 32 bits of a generic LDS pointer are the LDS byte offset
    unsigned l = (unsigned)(unsigned long long)lds_dst;
    asm volatile("global_load_async_to_lds_b128 %0, %1, off"
                 :: "v"(l), "v"(gsrc) : "memory");
}
__device__ __forceinline__ void wait_async_le4() {
    asm volatile("s_wait_asynccnt 0x4" ::: "memory");
}
__device__ __forceinline__ void wait_async_0() {
    asm volatile("s_wait_asynccnt 0x0" ::: "memory");
}

// Stage one 16-deep K chunk of both operands in native layout (4 asyncs/thread).
// TRANSA==0: sA[m][k] = A[(bm+m)*512+kb+k], sB[n][k] = B[(bn+n)*512+kb+k] (stride 20)
// TRANSA==1: sA[k][m] = A[(kb+k)*512+bm+m], sB[k][n] = B[(kb+k)*512+bn+n] (stride 128)
template <int TRANSA>
__device__ __forceinline__ void stage_async(const float* __restrict__ Az,
                                            const float* __restrict__ Bz,
                                            float* sA, float* sB,
                                            int bm, int bn, int kb, int tid)
{
    if (TRANSA == 0) {
        const int r  = tid >> 2;           // 0..63
        const int k4 = (tid & 3) * 4;      // 0,4,8,12
        async_b128(&sA[(r     ) * LDSTRIDE + k4], Az + (size_t)(bm + r     ) * NMAT + kb + k4);
        async_b128(&sA[(r + 64) * LDSTRIDE + k4], Az + (size_t)(bm + r + 64) * NMAT + kb + k4);
        async_b128(&sB[(r     ) * LDSTRIDE + k4], Bz + (size_t)(bn + r     ) * NMAT + kb + k4);
        async_b128(&sB[(r + 64) * LDSTRIDE + k4], Bz + (size_t)(bn + r + 64) * NMAT + kb + k4);
    } else {
        const int k  = tid >> 4;           // 0..15
        const int m8 = (tid & 15) * 8;     // 0..120
        async_b128(&sA[k * 128 + m8    ], Az + (size_t)(kb + k) * NMAT + bm + m8);
        async_b128(&sA[k * 128 + m8 + 4], Az + (size_t)(kb + k) * NMAT + bm + m8 + 4);
        async_b128(&sB[k * 128 + m8    ], Bz + (size_t)(kb + k) * NMAT + bn + m8);
        async_b128(&sB[k * 128 + m8 + 4], Bz + (size_t)(kb + k) * NMAT + bn + m8 + 4);
    }
}

// ---------------------------------------------------------------------------
// Tiled GEMM, M=N=K=512, batched over blockIdx.z.
// TRANSA==0 ("NT"):  C[m,n] = sum_k A[m*512+k] * Bm[n*512+k]
// TRANSA==1 ("TN"):  C[m,n] = sum_k A[k*512+m] * Bm[k*512+n]   (for v5^T @ v7)
// Epilogue: r = acc*scale; if(bias) r += bias[n]; if(E) r *= E[m,n];
//           if(R) r += R[m,n]; if(accum) C += r else C = r.
// Block: 256 threads = 8 wave32s; block tile 128x128; wave tile 32x64
// (2x4 WMMA 16x16 tiles); double-buffered async LDS staging, K chunks of 16.
// ---------------------------------------------------------------------------
template <int TRANSA>
__global__ __launch_bounds__(256)
void gemm512_wmma(const float* __restrict__ A,      // batched (z*MATELEMS)
                  const float* __restrict__ Bmat,   // batched iff batchedB
                  const float* __restrict__ bias,   // [512] or null
                  const float* __restrict__ Emul,   // batched or null
                  const float* __restrict__ Radd,   // batched or null
                  float* __restrict__ C,            // batched
                  float scale, int accum, int batchedB)
{
    __shared__ float sAbuf[2][128 * LDSTRIDE];   // 10 KB each
    __shared__ float sBbuf[2][128 * LDSTRIDE];

    const int bm   = blockIdx.y * 128;
    const int bn   = blockIdx.x * 128;
    const int z    = blockIdx.z;
    const int tid  = threadIdx.x;
    const int wid  = tid >> 5;
    const int lane = tid & 31;
    const int half = lane >> 4;        // A/B/C fragment lane-half (K or M split)
    const int l16  = lane & 15;
    const int wm   = (wid & 3) * 32;   // wave row offset in block tile (4 waves)
    const int wn   = (wid >> 2) * 64;  // wave col offset in block tile (2 waves)

    const float* Az = A + (size_t)z * MATELEMS;
    const float* Bz = Bmat + (batchedB ? (size_t)z * MATELEMS : 0);

    v8f acc[2][4];
    #pragma unroll
    for (int tm = 0; tm < 2; ++tm)
        #pragma unroll
        for (int tn = 0; tn < 4; ++tn)
            acc[tm][tn] = (v8f){0.f, 0.f, 0.f, 0.f, 0.f, 0.f, 0.f, 0.f};

    // software pipeline: chunk 0 in flight before the loop
    stage_async<TRANSA>(Az, Bz, sAbuf[0], sBbuf[0], bm, bn, 0, tid);

    for (int i = 0; i < NMAT / 16; ++i) {
        const int cur = i & 1;
        if (i + 1 < NMAT / 16) {
            stage_async<TRANSA>(Az, Bz, sAbuf[cur ^ 1], sBbuf[cur ^ 1],
                                bm, bn, (i + 1) * 16, tid);
            wait_async_le4();          // chunk i landed; chunk i+1 (4 ops) in flight
        } else {
            wait_async_0();
        }
        __syncthreads();               // all waves' async data visible

        const float* sA = sAbuf[cur];
        const float* sB = sBbuf[cur];
        #pragma unroll
        for (int ks = 0; ks < 4; ++ks) {
            const int kk = ks * 4 + half * 2;    // per-lane K base of the frag
            v2f a[2], b[4];
            #pragma unroll
            for (int tm = 0; tm < 2; ++tm) {
                const int m = wm + tm * 16 + l16;
                if (TRANSA == 0) {
                    a[tm].x = sA[m * LDSTRIDE + kk];
                    a[tm].y = sA[m * LDSTRIDE + kk + 1];
                } else {
                    a[tm].x = sA[kk * 128 + m];
                    a[tm].y = sA[(kk + 1) * 128 + m];
                }
            }
            #pragma unroll
            for (int tn = 0; tn < 4; ++tn) {
                const int n = wn + tn * 16 + l16;
                if (TRANSA == 0) {
                    b[tn].x = sB[n * LDSTRIDE + kk];
                    b[tn].y = sB[n * LDSTRIDE + kk + 1];
                } else {
                    b[tn].x = sB[kk * 128 + n];
                    b[tn].y = sB[(kk + 1) * 128 + n];
                }
            }
            #pragma unroll
            for (int tm = 0; tm < 2; ++tm)
                #pragma unroll
                for (int tn = 0; tn < 4; ++tn)
                    acc[tm][tn] = __builtin_amdgcn_wmma_f32_16x16x4_f32(
                        false, a[tm], false, b[tn],
                        (short)0, acc[tm][tn], false, false);
        }
        __syncthreads();               // compute done before buffer reuse
    }

    // ---- epilogue ----
    const float* Ez = Emul ? Emul + (size_t)z * MATELEMS : nullptr;
    const float* Rz = Radd ? Radd + (size_t)z * MATELEMS : nullptr;
    float* Cz = C + (size_t)z * MATELEMS;
    #pragma unroll
    for (int tm = 0; tm < 2; ++tm) {
        #pragma unroll
        for (int tn = 0; tn < 4; ++tn) {
            const int n_g = bn + wn + tn * 16 + l16;
            #pragma unroll
            for (int r = 0; r < 8; ++r) {
                const int m_g = bm + wm + tm * 16 + r + half * 8;
                const size_t idx = (size_t)m_g * NMAT + n_g;
                float v = acc[tm][tn][r] * scale;
                if (bias) v += bias[n_g];
                if (Ez)   v *= Ez[idx];
                if (Rz)   v += Rz[idx];
                if (accum) Cz[idx] += v;
                else       Cz[idx]  = v;
            }
        }
    }
}

// ---------------------------------------------------------------------------
// LayerNorm over last dim (512). One block (256 threads) per row; float2/lane.
// ---------------------------------------------------------------------------
__global__ __launch_bounds__(256)
void ln512_kernel(const float* __restrict__ in, const float* __restrict__ g,
                  const float* __restrict__ b, float* __restrict__ out)
{
    __shared__ float red[256];
    const int row = blockIdx.x;
    const int tid = threadIdx.x;
    const float2 xv = ((const float2*)(in + (size_t)row * NMAT))[tid];

    red[tid] = xv.x + xv.y;
    __syncthreads();
    #pragma unroll
    for (int off = 128; off > 0; off >>= 1) {
        if (tid < off) red[tid] += red[tid + off];
        __syncthreads();
    }
    const float mean = red[0] * (1.0f / 512.0f);
    __syncthreads();

    const float dx = xv.x - mean, dy = xv.y - mean;
    red[tid] = dx * dx + dy * dy;
    __syncthreads();
    #pragma unroll
    for (int off = 128; off > 0; off >>= 1) {
        if (tid < off) red[tid] += red[tid + off];
        __syncthreads();
    }
    const float rstd = rsqrtf(red[0] * (1.0f / 512.0f) + 1e-5f);

    const float2 gv = ((const float2*)g)[tid];
    const float2 bv = ((const float2*)b)[tid];
    float2 o;
    o.x = dx * rstd * gv.x + bv.x;
    o.y = dy * rstd * gv.y + bv.y;
    ((float2*)(out + (size_t)row * NMAT))[tid] = o;
}

// Wsum = Wq3 + Wk3 + Wv3 (folds v16+v17+v18 into a single GEMM)
__global__ void wsum_kernel(const float* __restrict__ a, const float* __restrict__ b,
                            const float* __restrict__ c, float* __restrict__ o)
{
    const int i = blockIdx.x * blockDim.x + threadIdx.x;
    if (i < MATELEMS) o[i] = a[i] + b[i] + c[i];
}

extern "C" void kernel_launch(void* const* d_in, const int* in_sizes, int n_in,
                              void* d_out, int out_size, void* d_ws, size_t ws_size,
                              hipStream_t stream)
{
    const float* x    = (const float*)d_in[0];
    const float* Wq1  = (const float*)d_in[1];
    const float* bq1  = (const float*)d_in[2];
    const float* Wk1  = (const float*)d_in[3];
    const float* bk1  = (const float*)d_in[4];
    const float* Wv1  = (const float*)d_in[5];
    const float* bv1  = (const float*)d_in[6];
    const float* Wq2  = (const float*)d_in[7];
    // d_in[8] = Wk2: computed-but-unused in the reference -> skipped
    const float* Wv2  = (const float*)d_in[9];
    const float* Wq3  = (const float*)d_in[10];
    const float* Wk3  = (const float*)d_in[11];
    const float* Wv3  = (const float*)d_in[12];
    const float* ln1g = (const float*)d_in[13];
    const float* ln1b = (const float*)d_in[14];
    const float* ln2g = (const float*)d_in[15];
    const float* ln2b = (const float*)d_in[16];
    float* out = (float*)d_out;

    // Workspace: 6 batched 512x512 fp32 buffers (32 MB each) + Wsum (1 MB)
    const size_t BATCH = (size_t)NBATCH * MATELEMS;
    float* B0 = (float*)d_ws;      // v1
    float* B1 = B0 + BATCH;        // v2
    float* B2 = B1 + BATCH;        // v3
    float* B3 = B2 + BATCH;        // t2 -> t3 -> F
    float* B4 = B3 + BATCH;        // v5 -> SUM
    float* B5 = B4 + BATCH;        // v7 -> v15
    float* Ws = B5 + BATCH;        // Wsum [512x512]

    const dim3 gG(4, 4, NBATCH);   // 512/128 n-tiles, 512/128 m-tiles, batch
    const dim3 bG(256);
    const float inv_sqrt = 0.04419417382415922f;   // 1/sqrt(512)

    wsum_kernel<<<MATELEMS / 256, 256, 0, stream>>>(Wq3, Wk3, Wv3, Ws);

    // v1, v2, v3 = x @ W^T + b
    gemm512_wmma<0><<<gG, bG, 0, stream>>>(x, Wq1, bq1, nullptr, nullptr, B0, 1.f, 0, 0);
    gemm512_wmma<0><<<gG, bG, 0, stream>>>(x, Wk1, bk1, nullptr, nullptr, B1, 1.f, 0, 0);
    gemm512_wmma<0><<<gG, bG, 0, stream>>>(x, Wv1, bv1, nullptr, nullptr, B2, 1.f, 0, 0);
    // t2 = (v1 @ v2^T) / sqrt(D)    (batched B operand)
    gemm512_wmma<0><<<gG, bG, 0, stream>>>(B0, B1, nullptr, nullptr, nullptr, B3, inv_sqrt, 0, 1);
    // v5 = t2 @ Wq2^T ; v7 = t2 @ Wv2^T
    gemm512_wmma<0><<<gG, bG, 0, stream>>>(B3, Wq2, nullptr, nullptr, nullptr, B4, 1.f, 0, 0);
    gemm512_wmma<0><<<gG, bG, 0, stream>>>(B3, Wv2, nullptr, nullptr, nullptr, B5, 1.f, 0, 0);
    // t3 = (v5^T @ v7) / sqrt(H)    (TN variant, both batched)
    gemm512_wmma<1><<<gG, bG, 0, stream>>>(B4, B5, nullptr, nullptr, nullptr, B3, inv_sqrt, 0, 1);
    // SUM = (t3@Wq3^T)*v1 + (t3@Wk3^T)*v3 + (t3@Wv3^T)*v2
    gemm512_wmma<0><<<gG, bG, 0, stream>>>(B3, Wq3, nullptr, B0, nullptr, B4, 1.f, 0, 0);
    gemm512_wmma<0><<<gG, bG, 0, stream>>>(B3, Wk3, nullptr, B2, nullptr, B4, 1.f, 1, 0);
    gemm512_wmma<0><<<gG, bG, 0, stream>>>(B3, Wv3, nullptr, B1, nullptr, B4, 1.f, 1, 0);
    // v15 = LN1(SUM)
    ln512_kernel<<<NBATCH * NMAT, 256, 0, stream>>>(B4, ln1g, ln1b, B5);
    // F = v15 @ Wsum^T + SUM   (== v16+v17+v18)
    gemm512_wmma<0><<<gG, bG, 0, stream>>>(B5, Ws, nullptr, nullptr, B4, B3, 1.f, 0, 0);
    // out = LN2(F)
    ln512_kernel<<<NBATCH * NMAT, 256, 0, stream>>>(B3, ln2g, ln2b, out);
}